// dvgo_36103495090512
// MI455X (gfx1250) — compile-verified
//
#include <hip/hip_runtime.h>
#include <math.h>

// DVGO render: trilinear density -> alpha, per-ray exclusive cumprod -> weights,
// trilinear rgb -> sigmoid -> segmented weighted sum. Gather bound; grids +
// streams (~150 MB) fit in MI455X's 192 MB L2, so default RT cache hints keep
// everything resident. No matrix contraction in this op -> WMMA inapplicable;
// the CDNA5 paths used are wave32 shuffles (ds), global f32 atomics, and
// global_prefetch_b8.

#define GSZ   160
#define GSZ2  (160 * 160)
#define GSZ3  (160 * 160 * 160)
#define NRAYS_DEFAULT 16384

__device__ __forceinline__ void grid_coord(float p, int& i0, float& f) {
    // align_corners=True: g = (p - (-1)) / 2 * (G-1), clamp to [0, G-1]
    float g = (p + 1.0f) * 0.5f * (float)(GSZ - 1);
    g = fminf(fmaxf(g, 0.0f), (float)(GSZ - 1));
    int i = (int)floorf(g);
    if (i > GSZ - 2) i = GSZ - 2;
    if (i < 0) i = 0;
    i0 = i;
    f  = g - (float)i;
}

__device__ __forceinline__ float trilerp(const float* __restrict__ g,
                                         int ix, int iy, int iz,
                                         float fx, float fy, float fz) {
    long base = ((long)ix * GSZ + iy) * GSZ + iz;
    float c000 = g[base],               c001 = g[base + 1];
    float c010 = g[base + GSZ],         c011 = g[base + GSZ + 1];
    float c100 = g[base + GSZ2],        c101 = g[base + GSZ2 + 1];
    float c110 = g[base + GSZ2 + GSZ],  c111 = g[base + GSZ2 + GSZ + 1];
    float c00 = fmaf(fz, c001 - c000, c000);
    float c01 = fmaf(fz, c011 - c010, c010);
    float c10 = fmaf(fz, c101 - c100, c100);
    float c11 = fmaf(fz, c111 - c110, c110);
    float c0  = fmaf(fy, c01 - c00, c00);
    float c1  = fmaf(fy, c11 - c10, c10);
    return fmaf(fx, c1 - c0, c0);
}

// ---------------- Pass 1: per-sample alpha (stored into weights slot) --------
__global__ void __launch_bounds__(256)
dvgo_pass1_alpha(const float* __restrict__ dens, const float* __restrict__ xyz,
                 float* __restrict__ alpha_out, int M) {
    int i = blockIdx.x * blockDim.x + threadIdx.x;
    if (i >= M) return;
    float x = xyz[3 * i + 0], y = xyz[3 * i + 1], z = xyz[3 * i + 2];
    int ix, iy, iz; float fx, fy, fz;
    grid_coord(x, ix, fx); grid_coord(y, iy, fy); grid_coord(z, iz, fz);
    float d  = trilerp(dens, ix, iy, iz, fx, fy, fz);
    // softplus (stable) with ACT_SHIFT=0, VOX_RATIO=1
    float sp = fmaxf(d, 0.0f) + log1pf(expf(-fabsf(d)));
    alpha_out[i] = 1.0f - expf(-sp);
}

// ---------------- Pass 2: one WAVE per ray, chunked wave32 product-scan ------
__global__ void __launch_bounds__(256)
dvgo_pass2_scan(const int* __restrict__ ray_id, float* __restrict__ w,
                float* __restrict__ ail, float* __restrict__ rgbm,
                int M, int R) {
    int wave = (int)((blockIdx.x * blockDim.x + threadIdx.x) >> 5);
    int lane = (int)(threadIdx.x & 31);
    if (wave >= R) return;
    const int r = wave;

    // Uniform binary searches (all lanes compute identically; sorted ray_id):
    // s = lower_bound(r), e = lower_bound(r+1)
    int s, e;
    {
        int lo = 0, hi = M;
        while (lo < hi) { int mid = (lo + hi) >> 1; if (ray_id[mid] < r) lo = mid + 1; else hi = mid; }
        s = lo;
        hi = M;
        while (lo < hi) { int mid = (lo + hi) >> 1; if (ray_id[mid] < r + 1) lo = mid + 1; else hi = mid; }
        e = lo;
    }

    float T = 1.0f;  // running transmittance carry across chunks
    for (int base = s; base < e; base += 32) {
        int i = base + lane;
        // Prefetch next chunk (gfx1250 global_prefetch_b8) while we scan this one.
        if (i + 32 < e) __builtin_prefetch(&w[i + 32], 0, 3);

        bool  valid = (i < e);
        float a  = valid ? w[i] : 0.0f;                     // alpha from pass 1
        float ac = fminf(fmaxf(a, 0.0f), 0.9999999f);       // clip(alpha, 0, 1-1e-7)
        float p  = valid ? (1.0f - ac) : 1.0f;

        // Wave32 inclusive multiplicative scan (Hillis-Steele)
        float incl = p;
        #pragma unroll
        for (int off = 1; off < 32; off <<= 1) {
            float v = __shfl_up(incl, off, 32);
            if (lane >= off) incl *= v;
        }
        float excl = __shfl_up(incl, 1, 32);
        if (lane == 0) excl = 1.0f;

        if (valid) w[i] = a * (T * excl);                   // weight = alpha * T_exclusive

        float tot = __shfl(incl, 31, 32);                   // product of whole chunk
        T *= tot;
    }

    if (lane == 0) {
        float last = (e > s) ? T : 1.0f;                    // alphainv_last
        ail[r] = last;
        // seed rgb_marched with + alphainv_last[:, None]
        rgbm[3 * r + 0] = last;
        rgbm[3 * r + 1] = last;
        rgbm[3 * r + 2] = last;
    }
}

// ---------------- Pass 3: rgb gather + wave32 segmented reduce + atomics -----
__global__ void __launch_bounds__(256)
dvgo_pass3_rgb(const float* __restrict__ k0, const float* __restrict__ xyz,
               const int* __restrict__ ray_id, const float* __restrict__ w,
               float* __restrict__ rgbm, int M) {
    int i    = blockIdx.x * blockDim.x + threadIdx.x;
    int lane = threadIdx.x & 31;                            // wave32
    int rid  = -1;
    float wr = 0.0f, wg = 0.0f, wb = 0.0f;
    if (i < M) {
        rid = ray_id[i];
        float wv = w[i];
        float x = xyz[3 * i + 0], y = xyz[3 * i + 1], z = xyz[3 * i + 2];
        int ix, iy, iz; float fx, fy, fz;
        grid_coord(x, ix, fx); grid_coord(y, iy, fy); grid_coord(z, iz, fz);
        float cr = trilerp(k0,            ix, iy, iz, fx, fy, fz);
        float cg = trilerp(k0 +     GSZ3, ix, iy, iz, fx, fy, fz);
        float cb = trilerp(k0 + 2 * GSZ3, ix, iy, iz, fx, fy, fz);
        wr = wv * (1.0f / (1.0f + expf(-cr)));
        wg = wv * (1.0f / (1.0f + expf(-cg)));
        wb = wv * (1.0f / (1.0f + expf(-cb)));
    }
    // Segmented suffix-sum over contiguous equal-rid runs (ray_id sorted, so
    // runs are contiguous inside a wave). After the loop, each lane holds the
    // sum from itself to the end of its run; the run head covers the full run.
    #pragma unroll
    for (int off = 1; off < 32; off <<= 1) {
        int   ro = __shfl_down(rid, off, 32);
        float ar = __shfl_down(wr,  off, 32);
        float ag = __shfl_down(wg,  off, 32);
        float ab = __shfl_down(wb,  off, 32);
        if ((lane + off < 32) && (ro == rid)) { wr += ar; wg += ag; wb += ab; }
    }
    int rprev = __shfl_up(rid, 1, 32);
    bool head = (lane == 0) || (rprev != rid);
    if (head && rid >= 0) {
        atomicAdd(&rgbm[3 * rid + 0], wr);
        atomicAdd(&rgbm[3 * rid + 1], wg);
        atomicAdd(&rgbm[3 * rid + 2], wb);
    }
}

extern "C" void kernel_launch(void* const* d_in, const int* in_sizes, int n_in,
                              void* d_out, int out_size, void* d_ws, size_t ws_size,
                              hipStream_t stream) {
    const float* dens   = (const float*)d_in[0];  // [1,160,160,160]
    const float* k0     = (const float*)d_in[1];  // [3,160,160,160]
    const float* xyz    = (const float*)d_in[2];  // [M,3]
    const int*   ray_id = (const int*)  d_in[3];  // [M], sorted

    const int M = in_sizes[3];
    int R = (out_size - M) / 4;                   // M + R + 3R total outputs
    if (R <= 0) R = NRAYS_DEFAULT;

    float* w_out = (float*)d_out;                 // [M]      weights
    float* ail   = w_out + M;                     // [R]      alphainv_last
    float* rgbm  = ail + R;                       // [R,3]    rgb_marched

    const int TB = 256;
    const int WAVES_PER_BLOCK = TB / 32;
    dvgo_pass1_alpha<<<(M + TB - 1) / TB, TB, 0, stream>>>(dens, xyz, w_out, M);
    dvgo_pass2_scan <<<(R + WAVES_PER_BLOCK - 1) / WAVES_PER_BLOCK, TB, 0, stream>>>(
        ray_id, w_out, ail, rgbm, M, R);
    dvgo_pass3_rgb  <<<(M + TB - 1) / TB, TB, 0, stream>>>(k0, xyz, ray_id, w_out, rgbm, M);
}